// CostVolumeConstructor_25194278158918
// MI455X (gfx1250) — compile-verified
//
#include <hip/hip_runtime.h>
#include <stdint.h>

// Problem shape (fixed by the reference harness)
#define NVIEW 3
#define NB    2
#define NC    32
#define NH    128
#define NW    160
#define ND    32
#define NHW   (NH * NW)
#define TPB   128     // 4 x wave32
#define DCHUNK 8      // depths per block: reuse LDS ref tile + rx 8x
#define EPSF  1e-6f

// ---------------- gfx1250 async global->LDS plumbing (guarded) --------------
#if defined(__has_builtin)
#if __has_builtin(__builtin_amdgcn_global_load_async_to_lds_b128)
#define HAVE_ASYNC_B128 1
#endif
#if __has_builtin(__builtin_amdgcn_global_load_async_to_lds_b32)
#define HAVE_ASYNC_B32 1
#endif
#if __has_builtin(__builtin_amdgcn_s_wait_asynccnt)
#define HAVE_WAIT_ASYNCCNT 1
#endif
#endif

// Builtin signatures (probe-confirmed via hipcc diagnostics):
//   b32 : (AS1 int*  src, AS3 int*  dst, imm offset, imm cpol)
//   b128: (AS1 int4* src, AS3 int4* dst, imm offset, imm cpol)
typedef int int4v __attribute__((__vector_size__(4 * sizeof(int))));
typedef __attribute__((address_space(3))) int   lds_int;
typedef __attribute__((address_space(1))) int   gbl_int;
typedef __attribute__((address_space(3))) int4v lds_int4;
typedef __attribute__((address_space(1))) int4v gbl_int4;

__device__ __forceinline__ lds_int* to_lds(void* p) {
  // AS3 pointers are 32-bit; generic LDS pointer's low 32 bits are the LDS
  // byte offset.
  return reinterpret_cast<lds_int*>((uint32_t)(uintptr_t)p);
}
__device__ __forceinline__ gbl_int* to_gbl(const void* p) {
  return reinterpret_cast<gbl_int*>((uintptr_t)p);  // also strips const
}
__device__ __forceinline__ lds_int4* to_lds4(void* p) {
  return reinterpret_cast<lds_int4*>((uint32_t)(uintptr_t)p);
}
__device__ __forceinline__ gbl_int4* to_gbl4(const void* p) {
  return reinterpret_cast<gbl_int4*>((uintptr_t)p);
}

__device__ __forceinline__ void wait_async_copies() {
#if defined(HAVE_WAIT_ASYNCCNT)
  __builtin_amdgcn_s_wait_asynccnt(0);
#elif defined(HAVE_ASYNC_B128) || defined(HAVE_ASYNC_B32)
  asm volatile("s_wait_asynccnt 0" ::: "memory");
#endif
}

__device__ __forceinline__ int iclamp(int v, int lo, int hi) {
  return v < lo ? lo : (v > hi ? hi : v);
}

// ---------------------------- Main cost-volume kernel -----------------------
// (defined first so the disasm snippet shows this kernel's async prologue)
__global__ __launch_bounds__(TPB) void cv_main(
    const float* __restrict__ feats,        // [V,B,C,H,W]
    const float* __restrict__ depth_hypos,  // [B,D]
    const float* __restrict__ rt,           // [(V-1)*B][12]
    float* __restrict__ out) {              // [B,C,D,H,W]
  __shared__ float s_ref[NC * TPB];

  const int tid = threadIdx.x;
  const int hw  = blockIdx.x * TPB + tid;   // HW = 20480 = 160 blocks exactly
  const int b   = blockIdx.z;
  const int d0  = blockIdx.y * DCHUNK;
  const int h   = hw / NW;
  const int w   = hw - h * NW;

  // Async DMA of the reference-view tile (32 channels x 128 pixels = 16 KB)
  // into LDS, c-major: s_ref[c*TPB + pixel]. Cooperative: the block moves
  // 1024 contiguous 16-byte chunks; 8 x global_load_async_to_lds_b128 per
  // thread. Overlaps with the homography setup below.
  {
    const float* grow = feats + (size_t)(0 * NB + b) * NC * NHW
                              + (size_t)blockIdx.x * TPB;  // tile row start
#if defined(HAVE_ASYNC_B128)
#pragma unroll
    for (int q = tid; q < NC * TPB / 4; q += TPB) {
      const int c   = q >> 5;        // 32 x b128 chunks per 128-px row
      const int off = (q & 31) * 4;  // float offset within the row
      __builtin_amdgcn_global_load_async_to_lds_b128(
          to_gbl4(grow + (size_t)c * NHW + off),
          to_lds4(&s_ref[c * TPB + off]), 0, 0);
    }
#elif defined(HAVE_ASYNC_B32)
#pragma unroll
    for (int c = 0; c < NC; ++c)
      __builtin_amdgcn_global_load_async_to_lds_b32(
          to_gbl(grow + (size_t)c * NHW + tid),
          to_lds(&s_ref[c * TPB + tid]), 0, 0);
#else
#pragma unroll
    for (int q = tid; q < NC * TPB; q += TPB)
      s_ref[q] = grow[(size_t)(q >> 7) * NHW + (q & 127)];
#endif
  }

  // Depth-invariant part of the warp: rx = R * (x, y, 1)   (per src view)
  const float fx = (float)w, fy = (float)h;
  float rx[2][3], tt[2][3];
#pragma unroll
  for (int j = 0; j < 2; ++j) {
    const float* o = rt + (j * NB + b) * 12;   // uniform -> scalar loads
#pragma unroll
    for (int k = 0; k < 3; ++k) {
      rx[j][k] = o[k * 3 + 0] * fx + o[k * 3 + 1] * fy + o[k * 3 + 2];
      tt[j][k] = o[9 + k];
    }
  }
  const float* f1 = feats + (size_t)(1 * NB + b) * NC * NHW;
  const float* f2 = feats + (size_t)(2 * NB + b) * NC * NHW;

  wait_async_copies();   // s_wait_asynccnt 0 (own wave's DMA done)
  __syncthreads();       // all waves' chunks visible in LDS

  for (int dd = 0; dd < DCHUNK; ++dd) {
    const float dep = depth_hypos[b * ND + d0 + dd];

    int   oo[2][4];
    float ww[2][4];
#pragma unroll
    for (int j = 0; j < 2; ++j) {
      const float pz = rx[j][2] * dep + tt[j][2] + EPSF;
      const float rz = 1.0f / pz;
      const float X  = (rx[j][0] * dep + tt[j][0]) * rz;
      const float Y  = (rx[j][1] * dep + tt[j][1]) * rz;

      const float x0f = floorf(X), y0f = floorf(Y);
      const float wx1 = X - x0f, wx0 = 1.0f - wx1;
      const float wy1 = Y - y0f, wy0 = 1.0f - wy1;

      // validity on floats: NaN/inf coords compare false -> zero weight
      const bool vx0 = (x0f >= 0.0f)        && (x0f <= (float)(NW - 1));
      const bool vx1 = (x0f + 1.0f >= 0.0f) && (x0f + 1.0f <= (float)(NW - 1));
      const bool vy0 = (y0f >= 0.0f)        && (y0f <= (float)(NH - 1));
      const bool vy1 = (y0f + 1.0f >= 0.0f) && (y0f + 1.0f <= (float)(NH - 1));

      const int xi0 = iclamp((int)x0f,     0, NW - 1);
      const int xi1 = iclamp((int)x0f + 1, 0, NW - 1);
      const int yi0 = iclamp((int)y0f,     0, NH - 1);
      const int yi1 = iclamp((int)y0f + 1, 0, NH - 1);

      oo[j][0] = yi0 * NW + xi0;  ww[j][0] = (vx0 && vy0) ? wx0 * wy0 : 0.0f;
      oo[j][1] = yi0 * NW + xi1;  ww[j][1] = (vx1 && vy0) ? wx1 * wy0 : 0.0f;
      oo[j][2] = yi1 * NW + xi0;  ww[j][2] = (vx0 && vy1) ? wx0 * wy1 : 0.0f;
      oo[j][3] = yi1 * NW + xi1;  ww[j][3] = (vx1 && vy1) ? wx1 * wy1 : 0.0f;
    }

    float* outp = out + ((size_t)(b * NC) * ND + (size_t)(d0 + dd)) * NHW + hw;

#pragma unroll 8   // 64 independent gathers in flight: hide L2 latency
    for (int c = 0; c < NC; ++c) {
      const float ref = s_ref[c * TPB + tid];           // ds_load from LDS
      const float* p1 = f1 + (size_t)c * NHW;
      const float* p2 = f2 + (size_t)c * NHW;

      const float s1 = p1[oo[0][0]] * ww[0][0] + p1[oo[0][1]] * ww[0][1] +
                       p1[oo[0][2]] * ww[0][2] + p1[oo[0][3]] * ww[0][3];
      const float s2 = p2[oo[1][0]] * ww[1][0] + p2[oo[1][1]] * ww[1][1] +
                       p2[oo[1][2]] * ww[1][2] + p2[oo[1][3]] * ww[1][3];

      const float sum  = ref + s1 + s2;
      const float sq   = ref * ref + s1 * s1 + s2 * s2;
      const float mean = sum * (1.0f / 3.0f);
      float cv = sq * (1.0f / 3.0f) - mean * mean;
      cv = fmaxf(cv, 0.0f);  // clip>=0; NaN -> 0 (matches clip+nan_to_num)

      outp[(size_t)c * (ND * NHW)] = cv;  // coalesced along w
    }
  }
}

// -------- Setup: rel = P_src * inv(P_ref), store R(3x3)+t(3) per (view,b) ---
__global__ void cv_proj_setup(const float* __restrict__ proj,
                              float* __restrict__ rt) {
  int tid = threadIdx.x;
  if (tid >= (NVIEW - 1) * NB) return;
  int j = tid / NB;  // src view index - 1  (0 -> view1, 1 -> view2)
  int b = tid % NB;

  float m[16], inv[16];
  const float* p0 = proj + (0 * NB + b) * 16;
#pragma unroll
  for (int i = 0; i < 16; ++i) m[i] = p0[i];

  // Adjugate-based 4x4 inverse (gluInvertMatrix form)
  inv[0]  =  m[5]*m[10]*m[15] - m[5]*m[11]*m[14] - m[9]*m[6]*m[15] + m[9]*m[7]*m[14] + m[13]*m[6]*m[11] - m[13]*m[7]*m[10];
  inv[4]  = -m[4]*m[10]*m[15] + m[4]*m[11]*m[14] + m[8]*m[6]*m[15] - m[8]*m[7]*m[14] - m[12]*m[6]*m[11] + m[12]*m[7]*m[10];
  inv[8]  =  m[4]*m[9]*m[15]  - m[4]*m[11]*m[13] - m[8]*m[5]*m[15] + m[8]*m[7]*m[13] + m[12]*m[5]*m[11] - m[12]*m[7]*m[9];
  inv[12] = -m[4]*m[9]*m[14]  + m[4]*m[10]*m[13] + m[8]*m[5]*m[14] - m[8]*m[6]*m[13] - m[12]*m[5]*m[10] + m[12]*m[6]*m[9];
  inv[1]  = -m[1]*m[10]*m[15] + m[1]*m[11]*m[14] + m[9]*m[2]*m[15] - m[9]*m[3]*m[14] - m[13]*m[2]*m[11] + m[13]*m[3]*m[10];
  inv[5]  =  m[0]*m[10]*m[15] - m[0]*m[11]*m[14] - m[8]*m[2]*m[15] + m[8]*m[3]*m[14] + m[12]*m[2]*m[11] - m[12]*m[3]*m[10];
  inv[9]  = -m[0]*m[9]*m[15]  + m[0]*m[11]*m[13] + m[8]*m[1]*m[15] - m[8]*m[3]*m[13] - m[12]*m[1]*m[11] + m[12]*m[3]*m[9];
  inv[13] =  m[0]*m[9]*m[14]  - m[0]*m[10]*m[13] - m[8]*m[1]*m[14] + m[8]*m[2]*m[13] + m[12]*m[1]*m[10] - m[12]*m[2]*m[9];
  inv[2]  =  m[1]*m[6]*m[15]  - m[1]*m[7]*m[14]  - m[5]*m[2]*m[15] + m[5]*m[3]*m[14] + m[13]*m[2]*m[7]  - m[13]*m[3]*m[6];
  inv[6]  = -m[0]*m[6]*m[15]  + m[0]*m[7]*m[14]  + m[4]*m[2]*m[15] - m[4]*m[3]*m[14] - m[12]*m[2]*m[7]  + m[12]*m[3]*m[6];
  inv[10] =  m[0]*m[5]*m[15]  - m[0]*m[7]*m[13]  - m[4]*m[1]*m[15] + m[4]*m[3]*m[13] + m[12]*m[1]*m[7]  - m[12]*m[3]*m[5];
  inv[14] = -m[0]*m[5]*m[14]  + m[0]*m[6]*m[13]  + m[4]*m[1]*m[14] - m[4]*m[2]*m[13] - m[12]*m[1]*m[6]  + m[12]*m[2]*m[5];
  inv[3]  = -m[1]*m[6]*m[11]  + m[1]*m[7]*m[10]  + m[5]*m[2]*m[11] - m[5]*m[3]*m[10] - m[9]*m[2]*m[7]   + m[9]*m[3]*m[6];
  inv[7]  =  m[0]*m[6]*m[11]  - m[0]*m[7]*m[10]  - m[4]*m[2]*m[11] + m[4]*m[3]*m[10] + m[8]*m[2]*m[7]   - m[8]*m[3]*m[6];
  inv[11] = -m[0]*m[5]*m[11]  + m[0]*m[7]*m[9]   + m[4]*m[1]*m[11] - m[4]*m[3]*m[9]  - m[8]*m[1]*m[7]   + m[8]*m[3]*m[5];
  inv[15] =  m[0]*m[5]*m[10]  - m[0]*m[6]*m[9]   - m[4]*m[1]*m[10] + m[4]*m[2]*m[9]  + m[8]*m[1]*m[6]   - m[8]*m[2]*m[5];

  float det  = m[0]*inv[0] + m[1]*inv[4] + m[2]*inv[8] + m[3]*inv[12];
  float rdet = 1.0f / det;
#pragma unroll
  for (int i = 0; i < 16; ++i) {
    float v = inv[i] * rdet;
    if (!(v == v)) v = 0.0f;                        // NaN -> 0
    else if (v >  3.0e38f) v =  3.4028235e38f;      // +inf -> big
    else if (v < -3.0e38f) v = -3.4028235e38f;      // -inf -> -big
    inv[i] = v;
  }

  const float* ps = proj + ((1 + j) * NB + b) * 16;
  float* o = rt + (j * NB + b) * 12;
#pragma unroll
  for (int r = 0; r < 3; ++r) {
    float c0 = 0.f, c1 = 0.f, c2 = 0.f, c3 = 0.f;
#pragma unroll
    for (int k = 0; k < 4; ++k) {
      float a = ps[r * 4 + k];
      c0 += a * inv[k * 4 + 0];
      c1 += a * inv[k * 4 + 1];
      c2 += a * inv[k * 4 + 2];
      c3 += a * inv[k * 4 + 3];
    }
    o[r * 3 + 0] = c0;  // R row-major
    o[r * 3 + 1] = c1;
    o[r * 3 + 2] = c2;
    o[9 + r]     = c3;  // t
  }
}

// ------------------------------- Launcher -----------------------------------
extern "C" void kernel_launch(void* const* d_in, const int* in_sizes, int n_in,
                              void* d_out, int out_size, void* d_ws,
                              size_t ws_size, hipStream_t stream) {
  (void)in_sizes; (void)n_in; (void)out_size; (void)ws_size;
  const float* feats = (const float*)d_in[0];  // [3,2,32,128,160]
  const float* proj  = (const float*)d_in[1];  // [3,2,4,4]
  const float* depth = (const float*)d_in[2];  // [2,32]
  float* out = (float*)d_out;                  // [2,32,32,128,160]
  float* rt  = (float*)d_ws;                   // 48 floats: R|t per (view,b)

  cv_proj_setup<<<1, 32, 0, stream>>>(proj, rt);

  dim3 grid(NHW / TPB, ND / DCHUNK, NB);       // (160, 4, 2) blocks
  cv_main<<<grid, TPB, 0, stream>>>(feats, depth, rt, out);
}